// PlasticFCNetwork_1168231105118
// MI455X (gfx1250) — compile-verified
//
#include <hip/hip_runtime.h>
#include <hip/hip_bf16.h>

// Problem dims (from reference): x[B,T,N], w/alpha[N,N], eta[1] -> y[B,T,N]
static constexpr int BATCH = 8;
static constexpr int TT    = 128;
static constexpr int NN    = 512;

typedef float v2f __attribute__((ext_vector_type(2)));
typedef float v8f __attribute__((ext_vector_type(8)));
typedef int   v2i __attribute__((ext_vector_type(2)));

// ---------------------------------------------------------------------------
// CDNA5 async global->LDS copy (GLOBAL_LOAD_ASYNC_TO_LDS_B64, ASYNCcnt).
// Probe-learned signature: (v2i addrspace(1)* gsrc, v2i addrspace(3)* ldsdst,
//                           int offset, int cpol)
// Guarded so the file still compiles if the builtin is absent.
// ---------------------------------------------------------------------------
#if defined(__HIP_DEVICE_COMPILE__) && \
    __has_builtin(__builtin_amdgcn_global_load_async_to_lds_b64)
#define HAS_ASYNC_LDS 1
typedef __attribute__((address_space(1))) v2i glb_v2i;
typedef __attribute__((address_space(3))) v2i lds_v2i;
static __device__ __forceinline__ void async_copy8(float* dst_lds,
                                                   const float* src_glb) {
    float* s = const_cast<float*>(src_glb);
    __builtin_amdgcn_global_load_async_to_lds_b64(
        (glb_v2i*)s, (lds_v2i*)dst_lds, /*offset=*/0, /*cpol=*/0);
}
#endif

static __device__ __forceinline__ void async_wait0() {
#if defined(__HIP_DEVICE_COMPILE__) && \
    __has_builtin(__builtin_amdgcn_s_wait_asynccnt)
    __builtin_amdgcn_s_wait_asynccnt(0);
#elif defined(__HIP_DEVICE_COMPILE__)
    asm volatile("s_wait_asynccnt 0" ::: "memory");
#endif
}

// ---------------------------------------------------------------------------
// Phase 1: XW = X[B*T, N] x W[N, N], f32 WMMA 16x16x4.
// One 16x16 output tile per wave; 8 waves (256 thr) per block; 2048 tiles
// -> 256 blocks. K-loop steps by 4 (native f32 WMMA K).
// ---------------------------------------------------------------------------
__global__ __launch_bounds__(256)
void xw_gemm_wmma(const float* __restrict__ X,
                  const float* __restrict__ W,
                  float* __restrict__ XW) {
    const int wave = threadIdx.x >> 5;
    const int lane = threadIdx.x & 31;
    const int tile = blockIdx.x * 8 + wave;     // [0, 2048)
    const int tm   = tile >> 5;                 // 64 M-tiles (B*T/16)
    const int tn   = tile & 31;                 // 32 N-tiles (N/16)
    const int m0   = tm << 4;
    const int n0   = tn << 4;

    const int half = lane >> 4;                 // which K-pair this lane holds
    const int mr   = lane & 15;                 // A row / B col for this lane

    v8f acc = {};
    const float* aptr = X + (m0 + mr) * NN + half * 2;      // 8B aligned
    const float* bptr = W + (half * 2) * NN + (n0 + mr);

    for (int k0 = 0; k0 < NN; k0 += 4) {
        v2f a = *(const v2f*)aptr;              // A[m][k..k+1]
        v2f b;
        b.x = bptr[0];                          // B[k  ][n]
        b.y = bptr[NN];                         // B[k+1][n]
        acc = __builtin_amdgcn_wmma_f32_16x16x4_f32(
            /*neg_a=*/false, a, /*neg_b=*/false, b,
            /*c_mod=*/(short)0, acc, /*reuse_a=*/false, /*reuse_b=*/false);
        aptr += 4;
        bptr += 4 * NN;
    }

#pragma unroll
    for (int r = 0; r < 8; ++r) {
        const int mrow = r + half * 8;
        XW[(m0 + mrow) * NN + (n0 + mr)] = acc[r];
    }
}

// ---------------------------------------------------------------------------
// Phase 2: sequential plastic scan. Grid: 128 blocks = 8 batches x 16 column
// blocks of 32. Block: 256 threads = 8 waves; wave w owns i-chunk
// [w*64, w*64+64), lane owns one column j. Each thread keeps its 64 alpha
// values and 64 hebb values in VGPRs for the entire T loop.
//
// x_t rows are double-buffered in LDS; the next row is DMA'd in with
// GLOBAL_LOAD_ASYNC_TO_LDS_B64 while the current step computes, then
// fenced with s_wait_asynccnt + barrier.
// ---------------------------------------------------------------------------
__global__ __launch_bounds__(256)
void plastic_scan(const float* __restrict__ X,
                  const float* __restrict__ A,
                  const float* __restrict__ eta_p,
                  const float* __restrict__ XW,
                  float* __restrict__ Y) {
    __shared__ float xbuf[2][NN];       // double-buffered x_t rows
    __shared__ float partial[8][32];    // per-chunk partial sums
    __shared__ float ybuf[32];          // y for this column block

    const int tid   = threadIdx.x;
    const int j     = tid & 31;         // column within block
    const int chunk = tid >> 5;         // i-chunk (== wave index)
    const int ibase = chunk * 64;

    const int b  = blockIdx.x >> 4;     // 8 batches
    const int jb = blockIdx.x & 15;     // 16 column blocks
    const int j0 = jb * 32;
    const int jg = j0 + j;              // global column

    const float eta = eta_p[0];
    const float om  = 1.0f - eta;

    // Preload alpha column slice into registers (coalesced across lanes).
    float areg[64];
    float hreg[64];
#pragma unroll
    for (int r = 0; r < 64; ++r) {
        areg[r] = A[(ibase + r) * NN + jg];
        hreg[r] = 0.0f;
    }

    const float* xrow  = X + (size_t)b * TT * NN;
    const float* xwrow = XW + (size_t)b * TT * NN + j0;
    float*       yrow  = Y + (size_t)b * TT * NN + j0;

    // Prologue: bring x row 0 into buffer 0.
#ifdef HAS_ASYNC_LDS
    async_copy8(&xbuf[0][2 * tid], xrow + 2 * tid);
    async_wait0();
#else
    xbuf[0][tid]       = xrow[tid];
    xbuf[0][tid + 256] = xrow[tid + 256];
#endif
    __syncthreads();

    for (int t = 0; t < TT; ++t) {
        const int   cur = t & 1;
        const int   nxt = cur ^ 1;
        const float* xb = xbuf[cur];

        // Kick off the DMA of x_{t+1} into the alternate buffer; it overlaps
        // the whole reduce/tanh/update of this step.
        if (t + 1 < TT) {
#ifdef HAS_ASYNC_LDS
            async_copy8(&xbuf[nxt][2 * tid], xrow + NN + 2 * tid);
#else
            xbuf[nxt][tid]       = xrow[NN + tid];
            xbuf[nxt][tid + 256] = xrow[NN + tid + 256];
#endif
        }
        const float xwval = (tid < 32) ? xwrow[tid] : 0.0f;

        // Reduction pass: s = sum over this chunk's 64 i-values.
        float s = 0.0f;
#pragma unroll
        for (int r4 = 0; r4 < 64; r4 += 4) {
            const float4 xv = *(const float4*)&xb[ibase + r4];
            s += xv.x * areg[r4 + 0] * hreg[r4 + 0];
            s += xv.y * areg[r4 + 1] * hreg[r4 + 1];
            s += xv.z * areg[r4 + 2] * hreg[r4 + 2];
            s += xv.w * areg[r4 + 3] * hreg[r4 + 3];
        }
        partial[chunk][j] = s;
        __syncthreads();

        if (tid < 32) {
            float acc = xwval;
#pragma unroll
            for (int c = 0; c < 8; ++c) acc += partial[c][tid];
            const float yv = tanhf(acc);
            ybuf[tid] = yv;
            yrow[tid] = yv;                     // write output y[b,t,j0+tid]
        }
        __syncthreads();

        // Hebbian update pass (register RMW).
        const float yv  = ybuf[j];
        const float eyv = eta * yv;
#pragma unroll
        for (int r4 = 0; r4 < 64; r4 += 4) {
            const float4 xv = *(const float4*)&xb[ibase + r4];
            hreg[r4 + 0] = om * hreg[r4 + 0] + eyv * xv.x;
            hreg[r4 + 1] = om * hreg[r4 + 1] + eyv * xv.y;
            hreg[r4 + 2] = om * hreg[r4 + 2] + eyv * xv.z;
            hreg[r4 + 3] = om * hreg[r4 + 3] + eyv * xv.w;
        }

        // Next-step buffer must be fully landed (each wave waits its own
        // async ops, barrier publishes across waves).
#ifdef HAS_ASYNC_LDS
        async_wait0();
#endif
        __syncthreads();

        xrow  += NN;
        xwrow += NN;
        yrow  += NN;
    }
}

extern "C" void kernel_launch(void* const* d_in, const int* in_sizes, int n_in,
                              void* d_out, int out_size, void* d_ws, size_t ws_size,
                              hipStream_t stream) {
    const float* x     = (const float*)d_in[0];   // [B,T,N]
    const float* w     = (const float*)d_in[1];   // [N,N]
    const float* alpha = (const float*)d_in[2];   // [N,N]
    const float* eta   = (const float*)d_in[3];   // [1]
    float* out = (float*)d_out;                   // [B,T,N]
    float* xw  = (float*)d_ws;                    // [B*T, N] scratch (2 MB)

    // Phase 1: XW = X * W via f32 WMMA. 2048 tiles / 8 waves per block.
    xw_gemm_wmma<<<256, 256, 0, stream>>>(x, w, xw);

    // Phase 2: plastic scan, hebb + alpha resident in VGPRs per thread,
    // async double-buffered x rows.
    plastic_scan<<<BATCH * (NN / 32), 256, 0, stream>>>(x, alpha, eta, xw, out);
}